// PretrainSeq2SeqLossModel_72284299591968
// MI455X (gfx1250) — compile-verified
//
#include <hip/hip_runtime.h>
#include <math.h>

// Problem constants (from reference): B=32, L=1024, H=1024, C=30
constexpr int HDIM  = 1024;
constexpr int CDIM  = 30;
constexpr int MROWS = 32 * 1024;          // B*L
constexpr int IGNORE_INDEX = -100;

typedef float v2f __attribute__((ext_vector_type(2)));
typedef float v8f __attribute__((ext_vector_type(8)));

// ---------------------------------------------------------------------------
// Kernel 1: fused skinny GEMM (WMMA f32 16x16x4) + bias + softmax-CE per row.
// 256 threads = 8 waves; each wave computes a 16-row x 32-col logit tile.
// ---------------------------------------------------------------------------
__global__ __launch_bounds__(256) void fused_gemm_ce(
    const float* __restrict__ E,       // [MROWS, HDIM]
    const float* __restrict__ W,       // [CDIM, HDIM]
    const float* __restrict__ bias,    // [CDIM]
    const int*   __restrict__ targets, // [MROWS]
    float* __restrict__ preds,         // [MROWS] (argmax, stored as float)
    float* __restrict__ nll_ws)        // [MROWS] per-row nll (0 if masked)
{
    const int lane   = threadIdx.x & 31;
    const int wid    = threadIdx.x >> 5;            // wave in block: 0..7
    const int tile   = blockIdx.x * 8 + wid;        // 16-row tile index
    const int rowbase = tile * 16;

    // Per-wave logit staging: 16 rows x 32 cols, stride 33 to avoid conflicts
    __shared__ float slog[8][16 * 33];

    const int m     = lane & 15;                    // A row within tile / col idx
    const int khalf = lane >> 4;                    // 0: K={0,1}, 1: K={2,3}

    // A: lane reads E[rowbase+m][k0 + 2*khalf + {0,1}]
    const float* Arow = E + (size_t)(rowbase + m) * HDIM + 2 * khalf;

    // B = W^T. B-frag element (k, n): lane = 16*(k>>1) + n, vgpr = k&1
    // -> lane reads W[n][k0 + 2*khalf + {0,1}]. Clamp OOB rows (cols masked later).
    const int   col0 = m;                           // 0..15  (< 30, always valid)
    const int   col1 = 16 + m;                      // 16..31 (>=30 masked)
    const float* Wp0 = W + (size_t)col0 * HDIM + 2 * khalf;
    const float* Wp1 = W + (size_t)((col1 < CDIM) ? col1 : 0) * HDIM + 2 * khalf;

    v8f acc0 = {0.f, 0.f, 0.f, 0.f, 0.f, 0.f, 0.f, 0.f};
    v8f acc1 = {0.f, 0.f, 0.f, 0.f, 0.f, 0.f, 0.f, 0.f};

    for (int k0 = 0; k0 < HDIM; k0 += 32) {
        if (k0 + 256 < HDIM)
            __builtin_prefetch(Arow + k0 + 256, 0, 0);   // global_prefetch_b8
#pragma unroll
        for (int kk = 0; kk < 32; kk += 4) {
            const int k = k0 + kk;
            v2f a  = *(const v2f*)(Arow + k);
            v2f b0 = *(const v2f*)(Wp0 + k);
            v2f b1 = *(const v2f*)(Wp1 + k);
            // D = A(16x4) * B(4x16) + C   -- v_wmma_f32_16x16x4_f32
            acc0 = __builtin_amdgcn_wmma_f32_16x16x4_f32(
                false, a, false, b0, (short)0, acc0, false, false);
            acc1 = __builtin_amdgcn_wmma_f32_16x16x4_f32(
                false, a, false, b1, (short)0, acc1, false, false);
        }
    }

    // C/D layout: VGPR r -> row (r + 8*(lane>=16)), col = lane&15 (+16 for acc1)
    const float b0v = bias[col0];
    const float b1v = (col1 < CDIM) ? bias[col1] : 0.f;
    const int   rhalf = (lane >> 4) * 8;

#pragma unroll
    for (int r = 0; r < 8; ++r) {
        const int row = rhalf + r;
        slog[wid][row * 33 + col0] = acc0[r] + b0v;
        slog[wid][row * 33 + col1] = (col1 < CDIM) ? (acc1[r] + b1v) : -INFINITY;
    }

    __syncthreads();

    // Per-row softmax-CE: lanes 0..15 each own one row (serial over C=30).
    if (lane < 16) {
        const int row  = lane;
        const int grow = rowbase + row;
        const float* lrow = &slog[wid][row * 33];

        float mx = -INFINITY;
        int   amax = 0;
#pragma unroll
        for (int c = 0; c < CDIM; ++c) {
            const float v = lrow[c];
            if (v > mx) { mx = v; amax = c; }   // strict '>' => first-max (jnp.argmax)
        }
        float s = 0.f;
#pragma unroll
        for (int c = 0; c < CDIM; ++c)
            s += expf(lrow[c] - mx);

        const int t = targets[grow];
        float nll = 0.f;
        if (t != IGNORE_INDEX) {
            const float xt = lrow[t];
            nll = logf(s) + mx - xt;
        }
        preds[grow]  = (float)amax;
        nll_ws[grow] = nll;
    }
}

// ---------------------------------------------------------------------------
// Kernel 2: deterministic single-block reduction: loss = sum(nll)/max(count,1)
// ---------------------------------------------------------------------------
__global__ __launch_bounds__(1024) void loss_reduce(
    const float* __restrict__ nll,
    const int*   __restrict__ targets,
    float* __restrict__ loss_out,
    int n)
{
    __shared__ float ss[1024];
    __shared__ int   sc[1024];
    const int tid = threadIdx.x;

    float s = 0.f;
    int   c = 0;
    for (int i = tid; i < n; i += 1024) {
        s += nll[i];
        c += (targets[i] != IGNORE_INDEX) ? 1 : 0;
    }
    ss[tid] = s;
    sc[tid] = c;
    __syncthreads();

    for (int off = 512; off > 0; off >>= 1) {
        if (tid < off) {
            ss[tid] += ss[tid + off];
            sc[tid] += sc[tid + off];
        }
        __syncthreads();
    }
    if (tid == 0) {
        const int cnt = (sc[0] > 1) ? sc[0] : 1;
        loss_out[0] = ss[0] / (float)cnt;
    }
}

// ---------------------------------------------------------------------------
extern "C" void kernel_launch(void* const* d_in, const int* in_sizes, int n_in,
                              void* d_out, int out_size, void* d_ws, size_t ws_size,
                              hipStream_t stream) {
    const float* E       = (const float*)d_in[0];   // [32,1024,1024]
    const float* W       = (const float*)d_in[1];   // [30,1024]
    const float* bias    = (const float*)d_in[2];   // [30]
    const int*   targets = (const int*)d_in[3];     // [32,1024]

    float* preds = (float*)d_out;                   // [32768] predictions
    float* loss  = preds + MROWS;                   // [1] loss
    float* nll   = (float*)d_ws;                    // [32768] scratch

    // 2048 row-tiles / 8 waves per block = 256 blocks of 256 threads
    fused_gemm_ce<<<dim3(MROWS / (16 * 8)), dim3(256), 0, stream>>>(
        E, W, bias, targets, preds, nll);

    loss_reduce<<<dim3(1), dim3(1024), 0, stream>>>(nll, targets, loss, MROWS);
}